// StockLoRA_58050777972930
// MI455X (gfx1250) — compile-verified
//
#include <hip/hip_runtime.h>

typedef __attribute__((ext_vector_type(2))) float v2f;
typedef __attribute__((ext_vector_type(4))) float v4f;
typedef __attribute__((ext_vector_type(8))) float v8f;

#define DIM   256
#define TTILE 16

// One wave (32 lanes) per (batch, 16-row tile of T).
// Phase A: r[16x2] = latent_tile[16x256] @ loraA[256x2]   (VALU FMAs + 1 shfl_xor)
// Phase B: out_tile[16x256] = r @ loraB^T + beta          (16x V_WMMA_F32_16X16X4_F32,
//                                                          beta carried in the C operand)
__global__ __launch_bounds__(32) void stock_lora_kernel(
    const float* __restrict__ latent,   // [B, T, D]
    const int*   __restrict__ idx,      // [B]
    const float* __restrict__ embedA,   // [S, D*2]  (d-major, rank interleaved)
    const float* __restrict__ embedB,   // [S, D*2]
    const float* __restrict__ beta,     // [S, D]
    float*       __restrict__ out,      // [B, T, D]
    int seqT)
{
    const int lane  = threadIdx.x;        // 0..31
    const int tTile = blockIdx.x;         // 0..seqT/TTILE-1
    const int b     = blockIdx.y;         // 0..B-1
    const int stock = idx[b];

    const int t0   = tTile * TTILE;
    const int row  = lane & 15;           // t-row within tile / d-column within chunk
    const int half = lane >> 4;           // 0: d<128, 1: d>=128

    const float* lrow = latent
        + ((size_t)b * seqT + (size_t)(t0 + row)) * DIM + half * (DIM / 2);
    const float* Arow = embedA
        + (size_t)stock * (DIM * 2) + (size_t)half * (DIM / 2) * 2;

    // Prefetch next tile's slice of latent into cache (global_prefetch_b8).
    if (t0 + TTILE < seqT) {
        __builtin_prefetch(lrow + (size_t)TTILE * DIM, 0, 0);
    }

    // ---- Phase A: rank-2 projection, reduced over this lane's 128 d's ----
    float r0 = 0.0f, r1 = 0.0f;
#pragma unroll 8
    for (int i = 0; i < (DIM / 2) / 4; ++i) {          // 32 iters x 4 d's
        v4f x   = *(const v4f*)(lrow + i * 4);
        v4f a01 = *(const v4f*)(Arow + i * 8);         // {A[d,0],A[d,1],A[d+1,0],A[d+1,1]}
        v4f a23 = *(const v4f*)(Arow + i * 8 + 4);     // {A[d+2,*],A[d+3,*]}
        r0 = fmaf(x.x, a01.x, r0);  r1 = fmaf(x.x, a01.y, r1);
        r0 = fmaf(x.y, a01.z, r0);  r1 = fmaf(x.y, a01.w, r1);
        r0 = fmaf(x.z, a23.x, r0);  r1 = fmaf(x.z, a23.y, r1);
        r0 = fmaf(x.w, a23.z, r0);  r1 = fmaf(x.w, a23.w, r1);
    }
    // Combine the two d-halves: lane L += lane L^16 (full r for row = lane&15).
    r0 += __shfl_xor(r0, 16, 32);
    r1 += __shfl_xor(r1, 16, 32);

    // A-matrix [16x4] f32 layout: VGPR0 = K0 (lanes 0-15) / K2 (lanes 16-31),
    //                             VGPR1 = K1 / K3.  Rank 2 -> K2,K3 are zero.
    const bool lo16 = (lane < 16);
    v2f amat;
    amat.x = lo16 ? r0 : 0.0f;
    amat.y = lo16 ? r1 : 0.0f;

    const float* Bbase   = embedB + (size_t)stock * (DIM * 2);
    const float* betaRow = beta   + (size_t)stock * DIM;
    float*       orow    = out + ((size_t)b * seqT + (size_t)t0) * DIM;

    // ---- Phase B: out[16 x 16] = r[16x2] @ B^T[2x16] + beta, per d-chunk ----
    // Branch-free: ALL lanes load B (lanes 16-31 duplicate 0-15, L0 broadcast),
    // then cndmask zeroes the K=2/3 padding — no exec-mask divergence.
#pragma unroll
    for (int n = 0; n < DIM / 16; ++n) {
        const int dbase = n * 16;

        // B-matrix [4x16] f32: VGPR0 = row K0 (lanes 0-15) / row K2 (lanes 16-31),
        //                      VGPR1 = row K1 / row K3.  N = lane&15.
        v2f bv = *(const v2f*)(Bbase + (size_t)(dbase + row) * 2);
        v2f bmat;
        bmat.x = lo16 ? bv.x : 0.0f;    // K=0 row (or K=2 pad)
        bmat.y = lo16 ? bv.y : 0.0f;    // K=1 row (or K=3 pad)

        // C = beta broadcast down M: lane's column is dbase+row for all 8 VGPRs.
        const float bval = betaRow[dbase + row];
        v8f c;
#pragma unroll
        for (int j = 0; j < 8; ++j) c[j] = bval;

        v8f dacc = __builtin_amdgcn_wmma_f32_16x16x4_f32(
            /*neg_a=*/false, amat, /*neg_b=*/false, bmat,
            /*c_mod=*/(short)0, c, /*reuse_a=*/false, /*reuse_b=*/false);

        // D layout: VGPR j -> M = j (lanes 0-15) or j+8 (lanes 16-31); N = lane&15.
#pragma unroll
        for (int j = 0; j < 8; ++j) {
            const int m = j + half * 8;
            orow[(size_t)m * DIM + dbase + row] = dacc[j];
        }
    }
}

extern "C" void kernel_launch(void* const* d_in, const int* in_sizes, int n_in,
                              void* d_out, int out_size, void* d_ws, size_t ws_size,
                              hipStream_t stream) {
    const float* latent = (const float*)d_in[0];
    const int*   idx    = (const int*)  d_in[1];
    const float* embedA = (const float*)d_in[2];
    const float* embedB = (const float*)d_in[3];
    const float* betaP  = (const float*)d_in[4];
    float*       outP   = (float*)d_out;

    const int B    = in_sizes[1];                        // batch count
    const int seqT = (int)(in_sizes[0] / ((size_t)B * DIM));  // sequence length

    dim3 grid(seqT / TTILE, B);
    stock_lora_kernel<<<grid, 32, 0, stream>>>(latent, idx, embedA, embedB, betaP,
                                               outP, seqT);
}